// WaveRNN_83348135346765
// MI455X (gfx1250) — compile-verified
//
#include <hip/hip_runtime.h>
#include <stdint.h>

// ---------------------------------------------------------------------------
// WaveRNN on MI455X (gfx1250), compile-only target.
//
// 32 persistent workgroups = 4 batches x 8 slabs of 32 rows. Per step:
//  - async-copy slab+halo (34 rows) of h1 from L2-resident ping-pong buffer
//    into LDS (global_load_async_to_lds_b128 + s_wait_asynccnt),
//  - compute EDGE rows first, release-fence + arrive on per-batch counter,
//  - compute interior rows (hides inter-WGP barrier latency),
//  - spin-wait + acquire-fence, swap time levels.
// Coefficients: B == A-1 identity removes one field; {A,C} packed float2 in
// LDS (ds_load_b64). ~98KB LDS per WGP.
// ---------------------------------------------------------------------------

#define NXg       256
#define NYg       256
#define TSTEPS    512
#define NBATCH    4
#define SLABS     8
#define SLAB_ROWS 32                     // NXg / SLABS
#define NTHREADS  1024
#define SEGS      4                      // thread col-groups
#define SEG_ROWS  (SLAB_ROWS / SEGS)     // 8 rows per thread
#define CELLS     (SLAB_ROWS * NYg)      // 8192 cells per slab

// ---- gfx1250 async global->LDS copy (CDNA5 path) --------------------------
typedef int v4i __attribute__((ext_vector_type(4)));
typedef __attribute__((address_space(1))) v4i* g_v4i_ptr;   // global v4i*
typedef __attribute__((address_space(3))) v4i* l_v4i_ptr;   // LDS v4i*

__device__ __forceinline__ void async_g2l_b128(const void* gsrc, uint32_t lds_off) {
#if __has_builtin(__builtin_amdgcn_global_load_async_to_lds_b128)
  __builtin_amdgcn_global_load_async_to_lds_b128(
      (g_v4i_ptr)(uintptr_t)gsrc, (l_v4i_ptr)(uintptr_t)lds_off, 0, 0);
#else
  uint64_t ga = (uint64_t)(uintptr_t)gsrc;
  asm volatile("global_load_async_to_lds_b128 %0, %1, off"
               :: "v"(lds_off), "v"(ga) : "memory");
#endif
}

__device__ __forceinline__ void wait_async0() {
#if __has_builtin(__builtin_amdgcn_s_wait_asynccnt)
  __builtin_amdgcn_s_wait_asynccnt(0);
#else
  asm volatile("s_wait_asynccnt 0" ::: "memory");
#endif
}

// ---- per-batch inter-WGP barrier pieces (monotonic counter, agent scope) --
__device__ __forceinline__ void bar_arrive(unsigned* ctr) {
  // callers ensure a release fence + __syncthreads precede this
  if (threadIdx.x == 0) {
    __hip_atomic_fetch_add(ctr, 1u, __ATOMIC_RELAXED, __HIP_MEMORY_SCOPE_AGENT);
  }
}

__device__ __forceinline__ void bar_wait(unsigned* ctr, unsigned target) {
  if (threadIdx.x == 0) {
    while (__hip_atomic_load(ctr, __ATOMIC_RELAXED, __HIP_MEMORY_SCOPE_AGENT) < target) {
      __builtin_amdgcn_s_sleep(2);
    }
  }
  __syncthreads();
  __builtin_amdgcn_fence(__ATOMIC_ACQUIRE, "agent");   // invalidate stale WGP$
}

__global__ __launch_bounds__(NTHREADS, 1)
void wave_rnn_persistent(const float* __restrict__ xin,   // [B, T]
                         const float* __restrict__ cfld,  // [256,256]
                         const float* __restrict__ rfld,  // [256,256]
                         const float* __restrict__ bfld,  // [256,256]
                         float* __restrict__ out,         // [B, T, 2]
                         float* __restrict__ ubase,       // 2 x [B,256,256]
                         unsigned* __restrict__ ctrs) {   // [B] counters
  // LDS: field slab with halo rows + packed {A,C} coefficients (~98 KB)
  __shared__ __attribute__((aligned(16))) float  sH[(SLAB_ROWS + 2) * NYg];
  __shared__ __attribute__((aligned(16))) float2 sAC[CELLS];

  const int slab  = blockIdx.x;          // 0..7
  const int batch = blockIdx.y;          // 0..3
  const int tid   = threadIdx.x;
  const int r0    = slab * SLAB_ROWS;
  const int gbase = batch * (NXg * NYg) + r0 * NYg;

  const int seg = tid >> 8;              // 0..3 (8 rows each)
  const int col = tid & (NYg - 1);       // 0..255

  float* u0 = ubase;                      // buffer 0: [B,256,256]
  float* u1 = ubase + NBATCH * NXg * NYg; // buffer 1
  unsigned* ctr = ctrs + batch;

  // ---- one-time init --------------------------------------------------
  for (int i = tid; i < CELLS; i += NTHREADS) {   // h0 = 0, both levels
    u0[gbase + i] = 0.0f;
    u1[gbase + i] = 0.0f;
  }
  if (tid < NYg) {                                // Dirichlet halo rows
    sH[tid] = 0.0f;
    sH[(SLAB_ROWS + 1) * NYg + tid] = 0.0f;
  }
  // coefficients: A = 2*inv_dt2*inv_den ; C = c^2*rho*inv_den ; B == A-1
  const int cbase = r0 * NYg;
  for (int i = tid; i < CELLS; i += NTHREADS) {
    float bh      = 10.0f * bfld[cbase + i];      // b/(2*dt)
    float inv_den = 1.0f / (400.0f + bh);         // 1/(inv_dt2 + b_half)
    float cc      = cfld[cbase + i];
    sAC[i] = make_float2(800.0f * inv_den, cc * cc * rfld[cbase + i] * inv_den);
  }

  unsigned phase = 1;
  __builtin_amdgcn_fence(__ATOMIC_RELEASE, "agent");
  __syncthreads();
  bar_arrive(ctr);
  bar_wait(ctr, SLABS * phase); phase++;          // zeros + coefs visible

  const float* xsrc = xin + batch * TSTEPS;

  int cur = 0;
  for (int s = 0; s < TSTEPS; ++s) {
    float* ucur  = (cur == 0) ? u0 : u1;   // h1 (current)
    float* uprev = (cur == 0) ? u1 : u0;   // h2 (previous) -> becomes h_new

    // -- stage h1 slab + halo rows into LDS via async copies ------------
    const int gr0   = (r0 > 0) ? (r0 - 1) : 0;
    const int gr1   = (r0 + SLAB_ROWS + 1 < NXg) ? (r0 + SLAB_ROWS + 1) : NXg;
    const int slot0 = gr0 - (r0 - 1);                 // 0 or 1
    const int nch   = (gr1 - gr0) * (NYg / 4);        // 16B chunks
    const float* gsrc = ucur + batch * (NXg * NYg) + gr0 * NYg;
    const uint32_t lbase = (uint32_t)(uintptr_t)(&sH[slot0 * NYg]);
    for (int j = tid; j < nch; j += NTHREADS) {
      async_g2l_b128(gsrc + j * 4, lbase + (uint32_t)j * 16u);
    }
    wait_async0();
    __syncthreads();

    const float xi = xsrc[s];

    auto do_cell = [&](int lr) {
      const int hrow = lr + 1;                   // sH row (halo offset)
      const int i    = lr * NYg + col;
      const float ce = sH[hrow * NYg + col];
      const float up = sH[(hrow - 1) * NYg + col];
      const float dn = sH[(hrow + 1) * NYg + col];
      const float lf = (col > 0)       ? sH[hrow * NYg + col - 1] : 0.0f;
      const float rt = (col < NYg - 1) ? sH[hrow * NYg + col + 1] : 0.0f;
      const float lap = up + dn + lf + rt - 4.0f * ce;

      const int g     = gbase + i;
      const float h2  = uprev[g];
      const float2 ac = sAC[i];
      // A*h1 - (A-1)*h2 + C*lap  ==  fma(C, lap, fma(A, h1-h2, h2))
      float hnew = fmaf(ac.y, lap, fmaf(ac.x, ce - h2, h2));

      const int grow = r0 + lr;
      if (grow == 20 && col == 128) hnew += xi;        // point source
      uprev[g] = hnew;                                  // becomes new h1

      if (grow == 200 && (col == 100 || col == 156)) { // intensity probes
        out[batch * (TSTEPS * 2) + s * 2 + ((col == 100) ? 0 : 1)] = hnew * hnew;
      }
    };

    // Row order: seg0 ascending (row 0 first), seg3 descending (row 31
    // first) so both slab edge rows are finished in iteration 0.
    auto row_of = [&](int j) {
      return (seg == SEGS - 1) ? (SLAB_ROWS - 1 - j) : (seg * SEG_ROWS + j);
    };

    // -- phase 1: edge rows, then early barrier arrival -----------------
    do_cell(row_of(0));
    __builtin_amdgcn_fence(__ATOMIC_RELEASE, "agent"); // publish edge rows
    __syncthreads();
    bar_arrive(ctr);

    // -- phase 2: interior rows overlap the barrier latency --------------
#pragma unroll
    for (int j = 1; j < SEG_ROWS; ++j) do_cell(row_of(j));

    // own interior stores must be visible to our own async engine too
    __builtin_amdgcn_fence(__ATOMIC_RELEASE, "agent");
    bar_wait(ctr, SLABS * phase); phase++;             // all edges published
    cur ^= 1;                                          // swap time levels
  }
}

extern "C" void kernel_launch(void* const* d_in, const int* in_sizes, int n_in,
                              void* d_out, int out_size, void* d_ws, size_t ws_size,
                              hipStream_t stream) {
  (void)in_sizes; (void)n_in; (void)out_size; (void)ws_size;
  const float* x   = (const float*)d_in[0];
  const float* c   = (const float*)d_in[1];
  const float* rho = (const float*)d_in[2];
  const float* b   = (const float*)d_in[3];
  float* out = (float*)d_out;

  // ws layout: [0,256) barrier counters; [256, 256 + 2*4*65536*4) u buffers
  unsigned* ctrs = (unsigned*)d_ws;
  float* ubase   = (float*)((char*)d_ws + 256);

  (void)hipMemsetAsync(d_ws, 0, 256, stream);  // zero barrier counters

  dim3 grid(SLABS, NBATCH);
  wave_rnn_persistent<<<grid, NTHREADS, 0, stream>>>(x, c, rho, b, out, ubase, ctrs);
}